// GCN_41248865911240
// MI455X (gfx1250) — compile-verified
//
#include <hip/hip_runtime.h>
#include <hip/hip_bf16.h>

typedef __attribute__((ext_vector_type(2))) float v2f;
typedef __attribute__((ext_vector_type(8))) float v8f;

#define HDIM 64
#define GDIM 64
#define EPSV 1e-5f

// ---------------------------------------------------------------------------
// utility fills
// ---------------------------------------------------------------------------
__global__ void k_fill(float* __restrict__ p, long long n, float v) {
    long long i = (long long)blockIdx.x * blockDim.x + threadIdx.x;
    if (i < n) p[i] = v;
}

// degree count via atomics (deg pre-filled with 1.0 for the self loop)
__global__ void k_deg(float* __restrict__ deg, const int* __restrict__ col, int e) {
    int i = blockIdx.x * blockDim.x + threadIdx.x;
    if (i < e) atomicAdd(&deg[col[i]], 1.0f);
}

__global__ void k_rsqrt_inplace(float* __restrict__ d, int n) {
    int i = blockIdx.x * blockDim.x + threadIdx.x;
    if (i < n) { float x = d[i]; d[i] = (x > 0.0f) ? rsqrtf(x) : 0.0f; }
}

__global__ void k_norm(float* __restrict__ wnorm, const float* __restrict__ dinv,
                       const int* __restrict__ row, const int* __restrict__ col, int e) {
    int i = blockIdx.x * blockDim.x + threadIdx.x;
    if (i < e) wnorm[i] = dinv[row[i]] * dinv[col[i]];
}

// ---------------------------------------------------------------------------
// Dense transform: Hout[n][o] = sum_k X[n][k] * W[o][k]   (X:[n x 64], W:[64 x 64])
// fp32 WMMA 16x16x4. One wave computes a 16x64 output tile; 8 waves / block.
// W staged in LDS with row stride 66 floats (bank-conflict-free float2 reads).
// A frag (16x4 f32): lane L holds x[M0 + L%16][k + 2*(L/16) + {0,1}]  -> float2
// B frag (4x16 f32): lane L holds W[N0 + L%16][k + 2*(L/16) + {0,1}]  -> float2
// C/D (16x16 f32):   vgpr v, lane L -> row M0 + v + 8*(L/16), col N0 + L%16
// ---------------------------------------------------------------------------
__global__ void __launch_bounds__(256) k_gemm64(const float* __restrict__ X,
                                                const float* __restrict__ W,
                                                float* __restrict__ Hout, int n) {
    __shared__ float wlds[64 * 66];
    for (int i = threadIdx.x; i < 64 * 64; i += 256) {
        int r = i >> 6, c = i & 63;
        wlds[r * 66 + c] = W[i];
    }
    __syncthreads();

    const int wave = threadIdx.x >> 5;   // 0..7
    const int lane = threadIdx.x & 31;
    const int half = lane >> 4;          // 0/1 -> K pair select
    const int lm   = lane & 15;

    const int rowbase = (blockIdx.x * 8 + wave) * 16;
    int arow = rowbase + lm;
    if (arow >= n) arow = n - 1;         // clamp: WMMA needs full EXEC; tails masked at store
    const float* xrow = X + (size_t)arow * HDIM;

    v8f acc0 = {}, acc1 = {}, acc2 = {}, acc3 = {};

#pragma unroll
    for (int k = 0; k < 64; k += 4) {
        const int ko = k + 2 * half;
        v2f a  = *(const v2f*)(xrow + ko);
        v2f b0 = *(const v2f*)(&wlds[(0 * 16 + lm) * 66 + ko]);
        v2f b1 = *(const v2f*)(&wlds[(1 * 16 + lm) * 66 + ko]);
        v2f b2 = *(const v2f*)(&wlds[(2 * 16 + lm) * 66 + ko]);
        v2f b3 = *(const v2f*)(&wlds[(3 * 16 + lm) * 66 + ko]);
        acc0 = __builtin_amdgcn_wmma_f32_16x16x4_f32(false, a, false, b0, (short)0, acc0, false, false);
        acc1 = __builtin_amdgcn_wmma_f32_16x16x4_f32(false, a, false, b1, (short)0, acc1, false, false);
        acc2 = __builtin_amdgcn_wmma_f32_16x16x4_f32(false, a, false, b2, (short)0, acc2, false, false);
        acc3 = __builtin_amdgcn_wmma_f32_16x16x4_f32(false, a, false, b3, (short)0, acc3, false, false);
    }

#pragma unroll
    for (int v = 0; v < 8; ++v) {
        const int r = rowbase + v + 8 * half;
        if (r < n) {
            float* o = Hout + (size_t)r * HDIM + lm;
            o[0]  = acc0[v];
            o[16] = acc1[v];
            o[32] = acc2[v];
            o[48] = acc3[v];
        }
    }
}

// ---------------------------------------------------------------------------
// Edge scatter: agg[col] += h[row] * wnorm.  16 threads per edge -> the 256 B
// feature row is a coalesced read; 4 packed global_atomic_add_f32 per thread.
// ---------------------------------------------------------------------------
__global__ void __launch_bounds__(256) k_scatter(const float* __restrict__ h,
                                                 float* __restrict__ agg,
                                                 const int* __restrict__ row,
                                                 const int* __restrict__ col,
                                                 const float* __restrict__ wnorm, int e) {
    long long tid = (long long)blockIdx.x * blockDim.x + threadIdx.x;
    int eidx = (int)(tid >> 4);
    if (eidx >= e) return;
    const int c = ((int)tid & 15) * 4;
    const int r = row[eidx];
    const int t = col[eidx];
    const float w = wnorm[eidx];
    const float4 hv = *(const float4*)(h + (size_t)r * HDIM + c);
    float* dst = agg + (size_t)t * HDIM + c;
    atomicAdd(dst + 0, hv.x * w);
    atomicAdd(dst + 1, hv.y * w);
    atomicAdd(dst + 2, hv.z * w);
    atomicAdd(dst + 3, hv.w * w);
}

// ---------------------------------------------------------------------------
// Fused: add self-loop term + conv bias, eval-mode BN, ReLU.
// out[n][c] = relu((agg[n][c] + h[n][c]*dinv[n]^2 + bconv[c] - m[c])*g[c]*rsqrt(v[c]+eps) + bb[c])
// ---------------------------------------------------------------------------
__global__ void __launch_bounds__(256) k_selfbn(const float* __restrict__ agg,
                                                const float* __restrict__ hself,
                                                const float* __restrict__ dinv,
                                                const float* __restrict__ bconv,
                                                const float* __restrict__ gg,
                                                const float* __restrict__ bb,
                                                const float* __restrict__ mm,
                                                const float* __restrict__ vv,
                                                float* __restrict__ out, int n) {
    long long tid = (long long)blockIdx.x * blockDim.x + threadIdx.x;
    const int node = (int)(tid >> 4);
    if (node >= n) return;
    const int c = ((int)tid & 15) * 4;
    const float di = dinv[node];
    const float self = di * di;
    const float4 av = *(const float4*)(agg + (size_t)node * HDIM + c);
    const float4 hv = *(const float4*)(hself + (size_t)node * HDIM + c);
    float4 o;
    {
        const float s = gg[c + 0] * rsqrtf(vv[c + 0] + EPSV);
        o.x = fmaxf((av.x + hv.x * self + bconv[c + 0] - mm[c + 0]) * s + bb[c + 0], 0.0f);
    }
    {
        const float s = gg[c + 1] * rsqrtf(vv[c + 1] + EPSV);
        o.y = fmaxf((av.y + hv.y * self + bconv[c + 1] - mm[c + 1]) * s + bb[c + 1], 0.0f);
    }
    {
        const float s = gg[c + 2] * rsqrtf(vv[c + 2] + EPSV);
        o.z = fmaxf((av.z + hv.z * self + bconv[c + 2] - mm[c + 2]) * s + bb[c + 2], 0.0f);
    }
    {
        const float s = gg[c + 3] * rsqrtf(vv[c + 3] + EPSV);
        o.w = fmaxf((av.w + hv.w * self + bconv[c + 3] - mm[c + 3]) * s + bb[c + 3], 0.0f);
    }
    *(float4*)(out + (size_t)node * HDIM + c) = o;
}

// ---------------------------------------------------------------------------
// LDS-staged global mean pool accumulation: 2048 nodes per block into a
// 64x64 f32 LDS tile (ds_add_f32), then one flush of block partials.
// ---------------------------------------------------------------------------
__global__ void __launch_bounds__(256) k_pool(const float* __restrict__ h,
                                              const int* __restrict__ batch,
                                              float* __restrict__ pool,
                                              float* __restrict__ cnt, int n) {
    __shared__ float lsum[GDIM * HDIM];
    __shared__ float lcnt[GDIM];
    for (int i = threadIdx.x; i < GDIM * HDIM; i += 256) lsum[i] = 0.0f;
    if (threadIdx.x < GDIM) lcnt[threadIdx.x] = 0.0f;
    __syncthreads();

    const int base = blockIdx.x * 2048;
    int nn = n - base;
    if (nn > 2048) nn = 2048;
    if (nn > 0) {
        const int items = nn * 16;
        for (int it = threadIdx.x; it < items; it += 256) {
            const int node = base + (it >> 4);
            const int c = (it & 15) * 4;
            const int g = batch[node];
            const float4 hv = *(const float4*)(h + (size_t)node * HDIM + c);
            atomicAdd(&lsum[g * HDIM + c + 0], hv.x);
            atomicAdd(&lsum[g * HDIM + c + 1], hv.y);
            atomicAdd(&lsum[g * HDIM + c + 2], hv.z);
            atomicAdd(&lsum[g * HDIM + c + 3], hv.w);
            if ((it & 15) == 0) atomicAdd(&lcnt[g], 1.0f);
        }
    }
    __syncthreads();
    for (int i = threadIdx.x; i < GDIM * HDIM; i += 256) {
        const float s = lsum[i];
        if (s != 0.0f) atomicAdd(&pool[i], s);
    }
    if (threadIdx.x < GDIM) {
        const float s = lcnt[threadIdx.x];
        if (s != 0.0f) atomicAdd(&cnt[threadIdx.x], s);
    }
}

// ---------------------------------------------------------------------------
// MLP head, single block: mean -> lin1+relu (64x64 -> 64x32) -> lin2 (->64x1)
// ---------------------------------------------------------------------------
__global__ void __launch_bounds__(256) k_head(const float* __restrict__ pool,
                                              const float* __restrict__ cnt,
                                              const float* __restrict__ l1W,
                                              const float* __restrict__ l1b,
                                              const float* __restrict__ l2W,
                                              const float* __restrict__ l2b,
                                              float* __restrict__ out) {
    __shared__ float p[GDIM * HDIM];
    __shared__ float z[GDIM * 32];
    for (int i = threadIdx.x; i < GDIM * HDIM; i += 256) {
        const int g = i >> 6;
        p[i] = pool[i] / fmaxf(cnt[g], 1.0f);
    }
    __syncthreads();
    for (int i = threadIdx.x; i < GDIM * 32; i += 256) {
        const int g = i >> 5, j = i & 31;
        float s = l1b[j];
#pragma unroll
        for (int c = 0; c < HDIM; ++c) s += p[g * HDIM + c] * l1W[j * HDIM + c];
        z[i] = fmaxf(s, 0.0f);
    }
    __syncthreads();
    for (int g = threadIdx.x; g < GDIM; g += 256) {
        float s = l2b[0];
#pragma unroll
        for (int j = 0; j < 32; ++j) s += z[g * 32 + j] * l2W[j];
        out[g] = s;
    }
}

// ---------------------------------------------------------------------------
extern "C" void kernel_launch(void* const* d_in, const int* in_sizes, int n_in,
                              void* d_out, int out_size, void* d_ws, size_t ws_size,
                              hipStream_t stream) {
    const float* x     = (const float*)d_in[0];
    const int*   ei    = (const int*)d_in[1];
    const int*   batch = (const int*)d_in[2];
    const float* W1    = (const float*)d_in[3];
    const float* b1    = (const float*)d_in[4];
    const float* W2    = (const float*)d_in[5];
    const float* b2    = (const float*)d_in[6];
    const float* bn1_g = (const float*)d_in[7];
    const float* bn1_b = (const float*)d_in[8];
    const float* bn1_m = (const float*)d_in[9];
    const float* bn1_v = (const float*)d_in[10];
    const float* bn2_g = (const float*)d_in[11];
    const float* bn2_b = (const float*)d_in[12];
    const float* bn2_m = (const float*)d_in[13];
    const float* bn2_v = (const float*)d_in[14];
    const float* l1W   = (const float*)d_in[15];
    const float* l1b   = (const float*)d_in[16];
    const float* l2W   = (const float*)d_in[17];
    const float* l2b   = (const float*)d_in[18];
    float* out = (float*)d_out;

    const int n = in_sizes[0] / HDIM;   // 100000
    const int e = in_sizes[1] / 2;      // 1000000
    const int* row = ei;                // sources
    const int* col = ei + e;            // targets

    float* ws = (float*)d_ws;
    const size_t nh = (size_t)n * HDIM;
    const size_t o_dinv  = 0;
    const size_t o_wnorm = ((size_t)n + 255) & ~(size_t)255;
    const size_t o_h     = o_wnorm + (((size_t)e + 255) & ~(size_t)255);
    const size_t o_agg   = o_h + ((nh + 255) & ~(size_t)255);
    const size_t o_pool  = o_agg + ((nh + 255) & ~(size_t)255);
    const size_t o_cnt   = o_pool + GDIM * HDIM;

    float* dinv  = ws + o_dinv;
    float* wnorm = ws + o_wnorm;
    float* hbuf  = ws + o_h;
    float* abuf  = ws + o_agg;
    float* pool  = ws + o_pool;
    float* cnt   = ws + o_cnt;

    const int TPB = 256;
    const int gN    = (n + TPB - 1) / TPB;
    const int gE    = (e + TPB - 1) / TPB;
    const int gNH   = (int)((nh + TPB - 1) / TPB);
    const int gN16  = (int)(((size_t)n * 16 + TPB - 1) / TPB);
    const int gE16  = (int)(((size_t)e * 16 + TPB - 1) / TPB);
    const int gGemm = (n + 127) / 128;
    const int gPool = (n + 2047) / 2048;

    // --- normalization coefficients ---
    k_fill<<<gN, TPB, 0, stream>>>(dinv, n, 1.0f);                 // self-loop degree
    k_deg<<<gE, TPB, 0, stream>>>(dinv, col, e);
    k_rsqrt_inplace<<<gN, TPB, 0, stream>>>(dinv, n);
    k_norm<<<gE, TPB, 0, stream>>>(wnorm, dinv, row, col, e);

    // --- layer 1 ---
    k_gemm64<<<gGemm, TPB, 0, stream>>>(x, W1, hbuf, n);           // h = x @ W1.T  (WMMA)
    k_fill<<<gNH, TPB, 0, stream>>>(abuf, (long long)nh, 0.0f);
    k_scatter<<<gE16, TPB, 0, stream>>>(hbuf, abuf, row, col, wnorm, e);
    k_selfbn<<<gN16, TPB, 0, stream>>>(abuf, hbuf, dinv, b1,
                                       bn1_g, bn1_b, bn1_m, bn1_v, hbuf, n);

    // --- layer 2 ---
    k_gemm64<<<gGemm, TPB, 0, stream>>>(hbuf, W2, abuf, n);        // h2 = h @ W2.T (WMMA)
    k_fill<<<gNH, TPB, 0, stream>>>(hbuf, (long long)nh, 0.0f);    // hbuf becomes agg2
    k_scatter<<<gE16, TPB, 0, stream>>>(abuf, hbuf, row, col, wnorm, e);
    k_selfbn<<<gN16, TPB, 0, stream>>>(hbuf, abuf, dinv, b2,
                                       bn2_g, bn2_b, bn2_m, bn2_v, abuf, n);

    // --- pool + head ---
    k_fill<<<1, TPB, 0, stream>>>(pool, GDIM * HDIM + GDIM, 0.0f); // pool & cnt contiguous (4160 <= 16*256? no: use enough blocks)
    k_fill<<<(GDIM * HDIM + GDIM + TPB - 1) / TPB, TPB, 0, stream>>>(pool, GDIM * HDIM + GDIM, 0.0f);
    k_pool<<<gPool, TPB, 0, stream>>>(abuf, batch, pool, cnt, n);
    k_head<<<1, TPB, 0, stream>>>(pool, cnt, l1W, l1b, l2W, l2b, out);
}